// FwFMLayer_64793876628085
// MI455X (gfx1250) — compile-verified
//
#include <hip/hip_runtime.h>
#include <stdint.h>

typedef float v2f __attribute__((ext_vector_type(2)));
typedef float v8f __attribute__((ext_vector_type(8)));

#define B_TOTAL 16384
#define F 50
#define D 128
#define FP 64              // padded j-dimension (M), 4 tiles of 16
#define KP 52              // padded i-dimension (K), 13 steps of 4
#define NT (D / 16)        // 8 column tiles
#define MT (FP / 16)       // 4 row tiles
#define KS (KP / 4)        // 13 K-steps
#define WAVES_PER_BLOCK 4
#define BLOCK (WAVES_PER_BLOCK * 32)

// Async global->LDS copy of 16 bytes per lane (CDNA5 path, tracked by ASYNCcnt).
__device__ __forceinline__ void async_copy_b128(uint32_t lds_byte_addr, const float* gptr) {
    asm volatile("global_load_async_to_lds_b128 %0, %1, off"
                 :: "v"(lds_byte_addr), "v"(gptr)
                 : "memory");
}

__device__ __forceinline__ void wait_asynccnt0() {
    asm volatile("s_wait_asynccnt 0x0" ::: "memory");
}

__global__ __launch_bounds__(BLOCK)
void fwfm_wmma_kernel(const float* __restrict__ x,   // [B, F, D]
                      const float* __restrict__ w,   // [F, F]
                      float* __restrict__ out)       // [B]
{
    __shared__ float lds_w[FP * FP];                    // masked W, [i][j], zero-padded
    __shared__ float lds_x[WAVES_PER_BLOCK][KP * D];    // per-wave X slab, rows >=F zeroed

    const int tid  = threadIdx.x;
    const int wave = tid >> 5;
    const int lane = tid & 31;

    const int b = blockIdx.x * WAVES_PER_BLOCK + wave;
    const float* __restrict__ xb = x + (size_t)b * (F * D);
    float* __restrict__ xs = &lds_x[wave][0];

    // LDS byte address of this wave's X slab (addrspace(3) offset)
    const uint32_t xs_lds =
        (uint32_t)(uintptr_t)(__attribute__((address_space(3))) float*)xs;

    // ---- Async-stage this wave's X (50x128 f32) into LDS: 16B/lane/row ----
#pragma unroll 2
    for (int m = 0; m < F; ++m) {
        async_copy_b128(xs_lds + (uint32_t)(m * D + lane * 4) * 4u,
                        xb + m * D + lane * 4);
    }
    // zero the K-padding rows (50, 51)
    {
        const float4 z = make_float4(0.f, 0.f, 0.f, 0.f);
        *(float4*)(xs + 50 * D + lane * 4) = z;
        *(float4*)(xs + 51 * D + lane * 4) = z;
    }

    // ---- Stage strict-upper-masked W into LDS (block-cooperative) ----
    for (int idx = tid; idx < FP * FP; idx += BLOCK) {
        const int i = idx >> 6;
        const int j = idx & (FP - 1);
        float v = 0.0f;
        if (i < F && j < F && i < j) v = w[i * F + j];
        lds_w[idx] = v;
    }

    wait_asynccnt0();        // async X staging complete
    __syncthreads();         // W visible to all waves

    const int half = lane >> 4;    // 0: lanes 0-15, 1: lanes 16-31
    const int l16  = lane & 15;

    float partial = 0.0f;

    for (int nt = 0; nt < NT; ++nt) {
        const int n0 = nt * 16;

        // Preload all B-fragments for this column tile: B[k][n] = X[k][n0+n]
        // (4x16 f32 B-frag: lanes 0-15 hold K={0,1}, lanes 16-31 hold K={2,3})
        v2f bfrag[KS];
#pragma unroll
        for (int k = 0; k < KS; ++k) {
            const int kr = k * 4 + half * 2;
            v2f bf;
            bf.x = xs[(kr + 0) * D + n0 + l16];
            bf.y = xs[(kr + 1) * D + n0 + l16];
            bfrag[k] = bf;
        }

        for (int mt = 0; mt < MT; ++mt) {
            const int m0 = mt * 16;
            v8f acc = {0.f, 0.f, 0.f, 0.f, 0.f, 0.f, 0.f, 0.f};
#pragma unroll
            for (int k = 0; k < KS; ++k) {
                const int kr = k * 4 + half * 2;
                // A-frag (16x4 f32): A[m][k] = Wmask[i=kr][j=m0+l16]
                v2f af;
                af.x = lds_w[(kr + 0) * FP + m0 + l16];
                af.y = lds_w[(kr + 1) * FP + m0 + l16];
                acc = __builtin_amdgcn_wmma_f32_16x16x4_f32(
                          /*neg_a=*/false, af, /*neg_b=*/false, bfrag[k],
                          /*c_mod=*/(short)0, acc, /*reuse_a=*/false, /*reuse_b=*/false);
            }
            // out[b] += X[m][n] * Y[m][n]; D-tile layout: m = m0 + r + 8*half, n = n0 + l16
#pragma unroll
            for (int r = 0; r < 8; ++r) {
                const int m = m0 + r + 8 * half;
                if (m < F) partial += acc[r] * xs[m * D + n0 + l16];
            }
        }
    }

    // ---- wave32 reduction ----
#pragma unroll
    for (int off = 16; off > 0; off >>= 1)
        partial += __shfl_xor(partial, off, 32);
    if (lane == 0) out[b] = partial;
}

extern "C" void kernel_launch(void* const* d_in, const int* in_sizes, int n_in,
                              void* d_out, int out_size, void* d_ws, size_t ws_size,
                              hipStream_t stream) {
    const float* x = (const float*)d_in[0];   // [16384, 50, 128] f32
    const float* w = (const float*)d_in[1];   // [50, 50] f32
    float* out = (float*)d_out;               // [16384] f32

    const int blocks = B_TOTAL / WAVES_PER_BLOCK;  // 4096
    fwfm_wmma_kernel<<<blocks, BLOCK, 0, stream>>>(x, w, out);
}